// TransformedEmbedding_36515811950880
// MI455X (gfx1250) — compile-verified
//
#include <hip/hip_runtime.h>

typedef __attribute__((ext_vector_type(2))) float v2f;
typedef __attribute__((ext_vector_type(4))) float v4f;
typedef __attribute__((ext_vector_type(8))) float v8f;

#define D 128
#define BATCH 16384
#define LDS_STRIDE 132   // 128 + 4 pad -> conflict-free b64 reads (see analysis)

// ---------------------------------------------------------------------------
// Kernel 1: M[x,y] = sum_z T[x,y,z]*tv[z]   (rows 0..16383 of a 16512-row job)
//           c[y]   = sum_z B[y,z]*tv[z]     (rows 16384..16511)
// One wave32 per 128-float row: float4 per lane, 5-step shuffle reduction.
// ---------------------------------------------------------------------------
__global__ __launch_bounds__(256) void k_reduce_tv(
    const float* __restrict__ T, const float* __restrict__ Bm,
    const float* __restrict__ tv, float* __restrict__ M, float* __restrict__ c)
{
    const int lane = threadIdx.x & 31;
    const int wave = threadIdx.x >> 5;
    const int row  = blockIdx.x * 8 + wave;          // 0 .. 16511

    const float* src = (row < D * D) ? (T + (size_t)row * D)
                                     : (Bm + (size_t)(row - D * D) * D);

    const v4f* s4 = (const v4f*)src;
    const v4f* t4 = (const v4f*)tv;
    v4f a = s4[lane];
    v4f t = t4[lane];
    float acc = a.x * t.x + a.y * t.y + a.z * t.z + a.w * t.w;

    #pragma unroll
    for (int off = 16; off > 0; off >>= 1)
        acc += __shfl_down(acc, off, 32);

    if (lane == 0) {
        if (row < D * D) M[row] = acc;
        else             c[row - D * D] = acc;
    }
}

// ---------------------------------------------------------------------------
// Kernel 2: W2T[o,x] = sum_y M[x,y] * op_w[o,y]       (stored TRANSPOSED)
//           bias2[o] = sum_y c[y]   * op_w[o,y] + op_b[o]
// block = o (128 blocks), thread = x (128 threads): op_w row o staged in LDS,
// W2T row stores are coalesced. Tiny kernel (4 MFLOP), lives in L2.
// ---------------------------------------------------------------------------
__global__ __launch_bounds__(128) void k_fold_opw(
    const float* __restrict__ M, const float* __restrict__ c,
    const float* __restrict__ op_w, const float* __restrict__ op_b,
    float* __restrict__ W2T, float* __restrict__ bias2)
{
    __shared__ float srow[D];
    const int x = threadIdx.x;
    const int o = blockIdx.x;

    srow[x] = op_w[(size_t)o * D + x];
    __syncthreads();

    const float* mrow = M + (size_t)x * D;
    float s = 0.f;
    #pragma unroll 4
    for (int y = 0; y < D; ++y) s += mrow[y] * srow[y];
    W2T[(size_t)o * D + x] = s;

    if (x == 0) {
        float sb = op_b[o];
        #pragma unroll 4
        for (int y = 0; y < D; ++y) sb += c[y] * srow[y];
        bias2[o] = sb;
    }
}

// ---------------------------------------------------------------------------
// Kernel 3: out[b,o] = sum_x w_embed[idx[b],x] * W2[x,o] + bias2[o]
// 128 blocks x 256 threads (8 waves).
//
// W2T is staged into padded LDS with GLOBAL_LOAD_ASYNC_TO_LDS_B128 (memory ->
// LDS direct, ASYNCcnt-tracked, no VGPR round-trip), overlapping the random
// embedding-row gather. The LDS address operand is derived from the actual
// &sW2T[...] address (ptrtoint = capture) so the compiler knows the asm
// writes sW2T and cannot fold the B-fragment loads to poison; a dummy store
// into the padding lanes (k=128..131, never read) additionally guarantees
// sW2T has visible stores.
//
// Each wave owns a 16(batch) x 128(out) strip: 8 v_wmma_f32_16x16x4_f32
// accumulator tiles, K stepped by 4, bias fused via the WMMA C operand.
//
// Fragment layouts (ISA 7.12.2, wave32):
//   A 16x4 : lane L -> M = L&15, K = 2*(L>>4) + {0,1} in vgpr {0,1}
//   B 4x16 : lane L -> N = L&15, K = 2*(L>>4) + {0,1} in vgpr {0,1}
//   C 16x16: vgpr r, lane L -> M = r + 8*(L>>4), N = L&15
// ---------------------------------------------------------------------------
__global__ __launch_bounds__(256) void k_embed_gemm(
    const float* __restrict__ w_embed, const int* __restrict__ indices,
    const float* __restrict__ W2T, const float* __restrict__ bias2,
    float* __restrict__ out)
{
    // Only LDS object in this kernel -> allocation-relative offset 0, which is
    // what the async-to-LDS instruction's LDS address operand needs.
    __shared__ float sW2T[D * LDS_STRIDE];       // 67.5 KB (col-major, padded)

    const int tid   = threadIdx.x;
    const int lane  = tid & 31;
    const int wave  = tid >> 5;
    const int m0    = (blockIdx.x * 8 + wave) * 16;   // 16 batch rows per wave
    const int mrow  = lane & 15;                      // M (A/C) or N (B/C) index
    const int khalf = (lane >> 4) * 2;                // K sub-pair selector

    // Kick off the async W2T staging first: 4096 x 16B lane-copies,
    // memory -> LDS with no VGPR data. gi indexes float4 chunks of W2T
    // (row-major [o][x]): col = gi/32, k4 = (gi%32)*4.
    {
        #pragma unroll
        for (int i = 0; i < (D * D / 4) / 256; ++i) {
            const int gi  = tid + i * 256;
            const int col = gi >> 5;
            const int k4  = (gi & 31) << 2;
            // ptrtoint of the real LDS address: lo-32 bits == addrspace(3)
            // byte offset, and it captures sW2T so the asm may write it.
            const unsigned lds_off =
                (unsigned)(size_t)&sW2T[col * LDS_STRIDE + k4];
            const unsigned g_off = (unsigned)(gi * 16);
            asm volatile("global_load_async_to_lds_b128 %0, %1, %2"
                         :
                         : "v"(lds_off), "v"(g_off), "s"(W2T)
                         : "memory");
        }
    }

    // Visible dummy store into the padding lanes (never read): guarantees
    // sW2T has IR-level stores so its loads can't be optimized away.
    if (tid < D) sW2T[tid * LDS_STRIDE + D] = 0.f;

    // Gather this lane's embedding row (lanes L and L+16 share a row) and
    // prefetch it so the random-gather latency overlaps the async staging.
    const long long erow = (long long)indices[m0 + mrow];
    const float* arow = w_embed + erow * D;
    __builtin_prefetch(arow + (lane >> 4) * 64, 0, 0);

    // Init accumulators with the fused bias (same column for all 8 vgprs).
    v8f acc[8];
    #pragma unroll
    for (int t = 0; t < 8; ++t) {
        const float bv = bias2[t * 16 + mrow];
        acc[t] = (v8f){bv, bv, bv, bv, bv, bv, bv, bv};
    }

    // Own async copies done, then barrier for the other waves' copies.
    asm volatile("s_wait_asynccnt 0" ::: "memory");
    __syncthreads();

    for (int k0 = 0; k0 < D; k0 += 4) {
        // A fragment: 2 contiguous floats of this lane's embedding row.
        v2f a = *(const v2f*)(arow + k0 + khalf);

        #pragma unroll
        for (int t = 0; t < 8; ++t) {
            const int col = t * 16 + mrow;
            // B fragment: one aligned 8-byte LDS load (k, k+1 contiguous).
            v2f b = *(const v2f*)&sW2T[col * LDS_STRIDE + k0 + khalf];
            acc[t] = __builtin_amdgcn_wmma_f32_16x16x4_f32(
                /*neg_a=*/false, a, /*neg_b=*/false, b,
                /*c_mod=*/(short)0, acc[t],
                /*reuse_a=*/false, /*reuse_b=*/false);
        }
    }

    // Write the 16x128 strip.
    const int mbase = m0 + 8 * (lane >> 4);
    #pragma unroll
    for (int t = 0; t < 8; ++t) {
        const int col = t * 16 + mrow;
        #pragma unroll
        for (int r = 0; r < 8; ++r)
            out[(size_t)(mbase + r) * D + col] = acc[t][r];
    }
}

// ---------------------------------------------------------------------------
extern "C" void kernel_launch(void* const* d_in, const int* in_sizes, int n_in,
                              void* d_out, int out_size, void* d_ws, size_t ws_size,
                              hipStream_t stream) {
    const int*   indices = (const int*)  d_in[0];
    const float* w_embed = (const float*)d_in[1];
    const float* T       = (const float*)d_in[2];
    const float* Bm      = (const float*)d_in[3];
    const float* tv      = (const float*)d_in[4];
    const float* op_w    = (const float*)d_in[5];
    const float* op_b    = (const float*)d_in[6];
    float* out = (float*)d_out;

    float* ws    = (float*)d_ws;
    float* M     = ws;                 // 16384 floats
    float* c     = ws + D * D;         //   128 floats
    float* W2T   = ws + D * D + D;     // 16384 floats (transposed: [o][x])
    float* bias2 = ws + 2 * D * D + D; //   128 floats

    // 16512 rows (T's 16384 + B's 128), 8 rows per 256-thread block.
    k_reduce_tv<<<(D * D + D) / 8, 256, 0, stream>>>(T, Bm, tv, M, c);
    k_fold_opw<<<D, 128, 0, stream>>>(M, c, op_w, op_b, W2T, bias2);
    k_embed_gemm<<<BATCH / 128, 256, 0, stream>>>(w_embed, indices, W2T, bias2, out);
}